// WaveletAttention_22265110463189
// MI455X (gfx1250) — compile-verified
//
#include <hip/hip_runtime.h>
#include <math.h>

typedef float v2f __attribute__((ext_vector_type(2)));
typedef float v8f __attribute__((ext_vector_type(8)));

// D = A(16x4) * B(4x16) + C, fp32 exact.
#define WMMA_F32(a, b, c) \
  __builtin_amdgcn_wmma_f32_16x16x4_f32(false, (a), false, (b), (short)0, (c), false, false)

// ---------------------------------------------------------------------------
// C[M=8192 x 512] = A[8192 x 512] @ W[512 x 512] + bias[512], row-major fp32.
// One wave per 32x64 output tile (B fragments reused across 2 row tiles);
// grid = (M/32, 512/64).
// ---------------------------------------------------------------------------
__global__ __launch_bounds__(32) void gemm512_bias_kernel(
    const float* __restrict__ A, const float* __restrict__ W,
    const float* __restrict__ bias, float* __restrict__ C) {
  const int lane = threadIdx.x & 31;
  const int half = lane >> 4;   // 0: K lanes 0/1, 1: K lanes 2/3
  const int lm   = lane & 15;
  const int x0 = blockIdx.x * 32;
  const int n0 = blockIdx.y * 64;
  const int K = 512, N = 512;

  v8f acc0[4] = {};
  v8f acc1[4] = {};
  for (int k0 = 0; k0 < K; k0 += 4) {
    // A fragments: rows x0+lm and x0+16+lm, cols k0+half*2 .. +1 (contiguous)
    v2f a0, a1;
    const float* ap0 = A + (size_t)(x0 + lm) * K + (k0 + half * 2);
    const float* ap1 = ap0 + (size_t)16 * K;
    a0.x = ap0[0]; a0.y = ap0[1];
    a1.x = ap1[0]; a1.y = ap1[1];
    // B fragments: rows k0+half*2 .. +1, cols n0 + t*16 + lm (shared by both row tiles)
    const float* wp = W + (size_t)(k0 + half * 2) * N + n0 + lm;
#pragma unroll
    for (int t = 0; t < 4; ++t) {
      v2f b;
      b.x = wp[t * 16];
      b.y = wp[N + t * 16];
      acc0[t] = WMMA_F32(a0, b, acc0[t]);
      acc1[t] = WMMA_F32(a1, b, acc1[t]);
    }
  }
#pragma unroll
  for (int t = 0; t < 4; ++t) {
    const int n = n0 + t * 16 + lm;
    const float bv = bias[n];
#pragma unroll
    for (int v = 0; v < 8; ++v) {
      const int row = x0 + v + half * 8;   // C/D layout: rows v (lanes<16), v+8 (lanes>=16)
      C[(size_t)row * N + n]        = acc0[t][v] + bv;
      C[(size_t)(row + 16) * N + n] = acc1[t][v] + bv;
    }
  }
}

// ---------------------------------------------------------------------------
// Wavelet decomposition level: x (B,2*hn,64,8) -> d = xa@ec_d, s = xa@ec_s,
// xa = [x_even || x_odd] along last axis (16 wide). One thread per (b,y,c).
// ---------------------------------------------------------------------------
__global__ void wavelet_kernel(const float* __restrict__ x,
                               const float* __restrict__ ecd,
                               const float* __restrict__ ecs,
                               float* __restrict__ d, float* __restrict__ s,
                               int hn) {
  const int idx = blockIdx.x * blockDim.x + threadIdx.x;
  const int total = 4 * hn * 64;
  if (idx >= total) return;
  const int c = idx & 63;
  const int y = (idx >> 6) % hn;
  const int b = idx / (hn * 64);

  const float* xe = x + ((size_t)((size_t)b * 2 * hn + 2 * y) * 64 + c) * 8;
  const float* xo = xe + 512;  // next sequence row (64*8 floats)
  float xa[16];
#pragma unroll
  for (int j = 0; j < 8; ++j) { xa[j] = xe[j]; xa[8 + j] = xo[j]; }

  float* dp = d + ((size_t)((size_t)b * hn + y) * 64 + c) * 8;
  float* sp = s + ((size_t)((size_t)b * hn + y) * 64 + c) * 8;
#pragma unroll
  for (int kk = 0; kk < 8; ++kk) {
    float ad = 0.f, as = 0.f;
#pragma unroll
    for (int j = 0; j < 16; ++j) {
      ad += xa[j] * ecd[j * 8 + kk];
      as += xa[j] * ecs[j * 8 + kk];
    }
    dp[kk] = ad; sp[kk] = as;
  }
}

// ---------------------------------------------------------------------------
// Reconstruction level: w = [vf + Us_i || Ud_i] (16 wide),
// out[2y] = w@rc_e, out[2y+1] = w@rc_o.  One thread per (b,y,c).
// ---------------------------------------------------------------------------
__global__ void reconstruct_kernel(const float* __restrict__ vf,
                                   const float* __restrict__ Usi,
                                   const float* __restrict__ Udi,
                                   const float* __restrict__ rce,
                                   const float* __restrict__ rco,
                                   float* __restrict__ out, int m) {
  const int idx = blockIdx.x * blockDim.x + threadIdx.x;
  const int total = 4 * m * 64;
  if (idx >= total) return;
  const int c = idx & 63;
  const int y = (idx >> 6) % m;
  const int b = idx / (m * 64);

  const size_t off = ((size_t)((size_t)b * m + y) * 64 + c) * 8;
  const float* vp = vf + off;
  const float* up = Usi + off;
  const float* dp = Udi + off;
  float w[16];
#pragma unroll
  for (int j = 0; j < 8; ++j) { w[j] = vp[j] + up[j]; w[8 + j] = dp[j]; }

  float* oe = out + ((size_t)((size_t)b * 2 * m + 2 * y) * 64 + c) * 8;
  float* oo = oe + 512;
#pragma unroll
  for (int kk = 0; kk < 8; ++kk) {
    float ae = 0.f, ao = 0.f;
#pragma unroll
    for (int j = 0; j < 16; ++j) {
      ae += w[j] * rce[j * 8 + kk];
      ao += w[j] * rco[j * 8 + kk];
    }
    oe[kk] = ae; oo[kk] = ao;
  }
}

// ---------------------------------------------------------------------------
// Attention: O[b,x,e,h] = sum_y softmax_y(scale * sum_e Q[b,x,e,h]K[b,y,e,h]) * V[b,y,e,h]
// Tensor layout (B, m, C=64, H=8) fp32. One wave per (16-row x-tile, head, batch).
// Scores strip kept in LDS **column-major** sc[y*16 + row] so the P*V
// A-fragment reads (hot loop) are bank-conflict-free. m=8 via -inf padding.
// ---------------------------------------------------------------------------
__global__ __launch_bounds__(32) void attention_kernel(
    const float* __restrict__ Q, const float* __restrict__ Km,
    const float* __restrict__ V, float* __restrict__ O,
    int m, float scale) {
  __shared__ float sc[1024 * 16];  // 64 KB; m <= 1024; layout sc[y*16 + row]
  const int lane = threadIdx.x & 31;
  const int half = lane >> 4;
  const int lm   = lane & 15;
  const int h  = blockIdx.y;
  const int b  = blockIdx.z;
  const int x0 = blockIdx.x * 16;
  const int ms = (m + 15) & ~15;          // padded column count
  const size_t base = (size_t)b * m * 512;

  // Preload Q A-fragments over the 64-wide channel (reduction) axis.
  v2f qa[16];
#pragma unroll
  for (int kc = 0; kc < 16; ++kc) {
    const int x = x0 + lm;
    const int e = kc * 4 + half * 2;
    float q0 = 0.f, q1 = 0.f;
    if (x < m) {
      const float* qp = Q + base + ((size_t)x * 64 + e) * 8 + h;
      q0 = qp[0]; q1 = qp[8];
    }
    qa[kc].x = q0; qa[kc].y = q1;
  }

  // Phase 1: scores into LDS (column-major).
  for (int y0 = 0; y0 < ms; y0 += 16) {
    v8f s = {};
    const int y = y0 + lm;
    const bool yok = (y < m);
#pragma unroll
    for (int kc = 0; kc < 16; ++kc) {
      const int e = kc * 4 + half * 2;
      float k0v = 0.f, k1v = 0.f;
      if (yok) {
        const float* kp = Km + base + ((size_t)y * 64 + e) * 8 + h;
        k0v = kp[0]; k1v = kp[8];
      }
      v2f bf; bf.x = k0v; bf.y = k1v;
      s = WMMA_F32(qa[kc], bf, s);
    }
#pragma unroll
    for (int v = 0; v < 8; ++v) {
      sc[(size_t)y * 16 + (v + half * 8)] = yok ? s[v] * scale : -3.0e38f;
    }
  }
  __syncthreads();

  // Phase 2: row softmax (2 lanes per row; element (row, col) at sc[col*16+row]).
  {
    const int row   = lane >> 1;
    const int sub   = lane & 1;
    const int hcols = ms >> 1;
    float* rp = sc + (size_t)(sub * hcols) * 16 + row;
    float mx = -3.4e38f;
    for (int j = 0; j < hcols; ++j) mx = fmaxf(mx, rp[(size_t)j * 16]);
    mx = fmaxf(mx, __shfl_xor(mx, 1, 32));
    float sum = 0.f;
    for (int j = 0; j < hcols; ++j) {
      float e = expf(rp[(size_t)j * 16] - mx);
      rp[(size_t)j * 16] = e;
      sum += e;
    }
    sum += __shfl_xor(sum, 1, 32);
    const float inv = 1.f / sum;
    for (int j = 0; j < hcols; ++j) rp[(size_t)j * 16] *= inv;
  }
  __syncthreads();

  // Phase 3: O = P * V.  A-frags (P rows x, y-chunks of 4) from LDS,
  // conflict-free: 16 lanes read 16 consecutive floats of one column.
  v8f o[4] = {};
  for (int y0 = 0; y0 < ms; y0 += 4) {
    v2f a;
    a.x = sc[(size_t)(y0 + half * 2) * 16 + lm];
    a.y = sc[(size_t)(y0 + half * 2 + 1) * 16 + lm];
    const int yr = y0 + half * 2;
#pragma unroll
    for (int t = 0; t < 4; ++t) {
      const int e = t * 16 + lm;
      float b0 = 0.f, b1 = 0.f;
      if (yr < m)     b0 = V[base + ((size_t)yr * 64 + e) * 8 + h];
      if (yr + 1 < m) b1 = V[base + ((size_t)(yr + 1) * 64 + e) * 8 + h];
      v2f bf; bf.x = b0; bf.y = b1;
      o[t] = WMMA_F32(a, bf, o[t]);
    }
  }
#pragma unroll
  for (int t = 0; t < 4; ++t) {
    const int e = t * 16 + lm;
#pragma unroll
    for (int v = 0; v < 8; ++v) {
      const int x = x0 + v + half * 8;
      if (x < m) O[base + ((size_t)x * 64 + e) * 8 + h] = o[t][v];
    }
  }
}

// ---------------------------------------------------------------------------
// Host orchestration.
// ---------------------------------------------------------------------------
extern "C" void kernel_launch(void* const* d_in, const int* in_sizes, int n_in,
                              void* d_out, int out_size, void* d_ws, size_t ws_size,
                              hipStream_t stream) {
  (void)in_sizes; (void)n_in; (void)out_size; (void)ws_size;
  const float* q     = (const float*)d_in[0];
  const float* k     = (const float*)d_in[1];
  const float* v     = (const float*)d_in[2];
  const float* Lq_w  = (const float*)d_in[3];
  const float* Lq_b  = (const float*)d_in[4];
  const float* Lk_w  = (const float*)d_in[5];
  const float* Lk_b  = (const float*)d_in[6];
  const float* Lv_w  = (const float*)d_in[7];
  const float* Lv_b  = (const float*)d_in[8];
  const float* out_w = (const float*)d_in[9];
  const float* out_b = (const float*)d_in[10];
  const float* ec_s  = (const float*)d_in[11];
  const float* ec_d  = (const float*)d_in[12];
  const float* rc_e  = (const float*)d_in[13];
  const float* rc_o  = (const float*)d_in[14];

  const size_t FULL = (size_t)4 * 2048 * 512;   // (B, 2048, 512)
  const size_t HALF = FULL / 2;                 // (B, 1024, 512)
  const size_t ULEV = (size_t)4 * 2040 * 512;   // stacked levels 1024..8

  float* ws  = (float*)d_ws;
  float* qf0 = ws; ws += FULL;
  float* qf1 = ws; ws += HALF;
  float* kf0 = ws; ws += FULL;
  float* kf1 = ws; ws += HALF;
  float* vf0 = ws; ws += FULL;
  float* vf1 = ws; ws += HALF;
  float* dq  = ws; ws += HALF;
  float* dk  = ws; ws += HALF;
  float* dv  = ws; ws += HALF;
  float* Ud  = ws; ws += ULEV;
  float* Us  = ws; ws += ULEV;

  const dim3 gblk(32);
  const dim3 ggrd(8192 / 32, 512 / 64);
  gemm512_bias_kernel<<<ggrd, gblk, 0, stream>>>(q, Lq_w, Lq_b, qf0);
  gemm512_bias_kernel<<<ggrd, gblk, 0, stream>>>(k, Lk_w, Lk_b, kf0);
  gemm512_bias_kernel<<<ggrd, gblk, 0, stream>>>(v, Lv_w, Lv_b, vf0);

  static const int pref[8] = {0, 1024, 1536, 1792, 1920, 1984, 2016, 2032};
  const float scale = 0.125f;  // 1/sqrt(64), T = 1.0

  float *qc = qf0, *qn = qf1, *kc = kf0, *kn = kf1, *vc = vf0, *vn = vf1;
  for (int lev = 0; lev < 8; ++lev) {
    const int hn = 1024 >> lev;
    const int threads = 4 * hn * 64;
    const dim3 wg((threads + 255) / 256);
    wavelet_kernel<<<wg, 256, 0, stream>>>(qc, ec_d, ec_s, dq, qn, hn);
    wavelet_kernel<<<wg, 256, 0, stream>>>(kc, ec_d, ec_s, dk, kn, hn);
    wavelet_kernel<<<wg, 256, 0, stream>>>(vc, ec_d, ec_s, dv, vn, hn);

    const int xt = (hn + 15) / 16;
    float* Udl = Ud + (size_t)pref[lev] * 2048;  // pref rows * B(4) * 512
    float* Usl = Us + (size_t)pref[lev] * 2048;
    attention_kernel<<<dim3(xt, 8, 4), 32, 0, stream>>>(dq, dk, dv, Udl, hn, scale);
    attention_kernel<<<dim3(xt, 8, 4), 32, 0, stream>>>(qn, kn, vn, Usl, hn, scale);

    float* t;
    t = qc; qc = qn; qn = t;
    t = kc; kc = kn; kn = t;
    t = vc; vc = vn; vn = t;
  }

  // After 8 swaps, vc == vf0 and holds vf at 8 rows/batch.
  float* rcur = vc;
  float* rnxt = vn;
  for (int i = 7; i >= 0; --i) {
    const int mrows = 1024 >> i;
    const int threads = 4 * mrows * 64;
    const dim3 rg((threads + 255) / 256);
    reconstruct_kernel<<<rg, 256, 0, stream>>>(rcur,
                                               Us + (size_t)pref[i] * 2048,
                                               Ud + (size_t)pref[i] * 2048,
                                               rc_e, rc_o, rnxt, mrows);
    float* t = rcur; rcur = rnxt; rnxt = t;
  }

  // rcur holds (B, 2048, 512); final projection into d_out.
  gemm512_bias_kernel<<<ggrd, gblk, 0, stream>>>(rcur, out_w, out_b, (float*)d_out);
}